// SS2D_3762391351588
// MI455X (gfx1250) — compile-verified
//
#include <hip/hip_runtime.h>

// ---------------- problem constants ----------------
#define B_SZ    2
#define H_IMG   64
#define W_IMG   64
#define L_SEQ   (H_IMG * W_IMG)      // 4096
#define D_MODEL 96
#define D_INNER 192
#define D_STATE 16
#define DT_RANK 6
#define K_DIRS  4
#define C_DBL   38                   // DT_RANK + 2*D_STATE
#define C_PAD   48                   // padded to 3 WMMA row tiles
#define LN_EPS  1e-5f

typedef float v2f __attribute__((ext_vector_type(2)));
typedef float v8f __attribute__((ext_vector_type(8)));

#if __has_builtin(__builtin_amdgcn_s_wait_asynccnt)
#define WAIT_ASYNC(n) __builtin_amdgcn_s_wait_asynccnt(n)
#else
#define WAIT_ASYNC(n) asm volatile("s_wait_asynccnt %0" ::"i"(n) : "memory")
#endif

// position in the flattened H*W image of element `l` of direction-k's sequence
__device__ __forceinline__ int dir_pos(int k, int l) {
  switch (k) {
    case 0: return l;
    case 1: return L_SEQ - 1 - l;
    case 2: {                                  // snake, top-down
      int i = l >> 6, c = l & 63;
      if (i & 1) c = W_IMG - 1 - c;
      return (i << 6) + c;
    }
    default: {                                 // snake, bottom-up (H even -> row H-1 rightward)
      int s = l >> 6, c = l & 63;
      int i = H_IMG - 1 - s;
      if (s & 1) c = W_IMG - 1 - c;
      return (i << 6) + c;
    }
  }
}

// ---------------- K0: zero-pad x_proj_weight (4,38,192) -> (4,48,192) ----------------
__global__ void pad_w(const float* __restrict__ xpw, float* __restrict__ xpwp) {
  int idx = blockIdx.x * blockDim.x + threadIdx.x;
  if (idx >= K_DIRS * C_PAD * D_INNER) return;
  int dc = idx % D_INNER;
  int c = (idx / D_INNER) % C_PAD;
  int k = idx / (D_INNER * C_PAD);
  xpwp[idx] = (c < C_DBL) ? xpw[((size_t)k * C_DBL + c) * D_INNER + dc] : 0.f;
}

// ---------------- K1/K6: C(MxN) = A(MxK,row) * W(NxK,row)^T, fp32 WMMA ----------------
// one wave computes a 16 x (NT*16) strip: A fragment reused across NT WMMA issues
template <int NT>
__global__ void gemm_nt_wmma(const float* __restrict__ A, const float* __restrict__ W,
                             float* __restrict__ C, int M, int N, int K) {
  const int lane = threadIdx.x & 31;
  const int lo = lane & 15, hi = lane >> 4;
  const int tm = blockIdx.x;
  const int tn0 = blockIdx.y * NT;
  const float* ap = A + (size_t)(tm * 16 + lo) * K + 2 * hi;   // A[row][k0+2hi .. +1]
  const float* wp = W + (size_t)(tn0 * 16 + lo) * K + 2 * hi;  // B[k][col] = W[col][k]
  v8f acc[NT] = {};
  for (int k0 = 0; k0 < K; k0 += 4) {
    v2f a = *(const v2f*)(ap + k0);
#pragma unroll
    for (int t = 0; t < NT; ++t) {
      v2f b = *(const v2f*)(wp + (size_t)t * 16 * K + k0);
      acc[t] = __builtin_amdgcn_wmma_f32_16x16x4_f32(false, a, false, b, (short)0,
                                                     acc[t], false, false);
    }
  }
#pragma unroll
  for (int t = 0; t < NT; ++t) {
    float* cp = C + (size_t)(tm * 16 + 8 * hi) * N + (tn0 + t) * 16 + lo;
#pragma unroll
    for (int r = 0; r < 8; ++r) cp[(size_t)r * N] = acc[t][r];
  }
}

// ---------------- K2: depthwise 3x3 conv + SiLU ----------------
// d-fastest mapping: coalesced reads of xz and coalesced stores of xfT.
// writes xf (B,DI,L) for the scan's u-stream and xfT (B,L,DI) for the x_dbl GEMM.
__global__ void conv_silu(const float* __restrict__ xz, const float* __restrict__ cw,
                          const float* __restrict__ cb, float* __restrict__ xf,
                          float* __restrict__ xfT) {
  int idx = blockIdx.x * blockDim.x + threadIdx.x;
  if (idx >= B_SZ * D_INNER * L_SEQ) return;
  int d = idx % D_INNER;
  int l = (idx / D_INNER) & (L_SEQ - 1);
  int b = idx / (D_INNER * L_SEQ);
  int hh = l >> 6, ww = l & 63;
  float acc = cb[d];
#pragma unroll
  for (int i = 0; i < 3; ++i) {
    int y = hh + i - 1;
    if (y < 0 || y >= H_IMG) continue;
#pragma unroll
    for (int j = 0; j < 3; ++j) {
      int x = ww + j - 1;
      if (x < 0 || x >= W_IMG) continue;
      float v = xz[(size_t)(b * L_SEQ + (y << 6) + x) * (2 * D_INNER) + d];
      acc = fmaf(cw[d * 9 + i * 3 + j], v, acc);
    }
  }
  float s = acc / (1.f + __expf(-acc));
  xf[(size_t)(b * D_INNER + d) * L_SEQ + l] = s;
  xfT[((size_t)b * L_SEQ + l) * D_INNER + d] = s;
}

// ---------------- K3: x_dbl[b,k,l,c] = sum_d xpwp[k,c,d] * xfT[b,pos_k(l),d] ----------------
// A = padded weights (48x192, unconditional loads), B gathered from K-contiguous xfT.
// Output stored COLUMN-MAJOR (b,k,L,48) so the scan reads one contiguous 192B line per step.
__global__ void xdbl_wmma(const float* __restrict__ xfT, const float* __restrict__ xpwp,
                          float* __restrict__ xdbl) {
  const int lane = threadIdx.x & 31;
  const int lo = lane & 15, hi = lane >> 4;
  const int tn0 = blockIdx.x * 4;            // L/16 col tiles, 4 per wave
  const int tm = blockIdx.y;                 // 3 row tiles (C_PAD/16)
  const int bk = blockIdx.z;                 // b*4 + k
  const int b = bk >> 2, k = bk & 3;
  const float* Arow = xpwp + ((size_t)k * C_PAD + tm * 16 + lo) * D_INNER + 2 * hi;
  const float* bp[4];
#pragma unroll
  for (int t = 0; t < 4; ++t) {
    int pos = dir_pos(k, (tn0 + t) * 16 + lo);
    bp[t] = xfT + ((size_t)b * L_SEQ + pos) * D_INNER + 2 * hi;
  }
  v8f acc[4] = {};
  for (int k0 = 0; k0 < D_INNER; k0 += 4) {
    v2f a = *(const v2f*)(Arow + k0);
#pragma unroll
    for (int t = 0; t < 4; ++t) {
      v2f bb = *(const v2f*)(bp[t] + k0);
      acc[t] = __builtin_amdgcn_wmma_f32_16x16x4_f32(false, a, false, bb, (short)0,
                                                     acc[t], false, false);
    }
  }
#pragma unroll
  for (int t = 0; t < 4; ++t) {
    float* out = xdbl + ((size_t)bk * L_SEQ + (tn0 + t) * 16 + lo) * C_PAD
                      + tm * 16 + 8 * hi;                 // 8 consecutive floats
#pragma unroll
    for (int r = 0; r < 8; ++r) out[r] = acc[t][r];
  }
}

// ---------------- K4: selective scan. one wave = 2 channels x 16 states ----------------
// Operand stream (L x 48 floats) is double-buffered through LDS in 16-step chunks
// using CDNA5 async global->LDS b128 copies tracked by ASYNCcnt.
#define CHUNK     16
#define CHUNK_F   (CHUNK * C_PAD)            // 768 floats = 3072 bytes = 6 x b128/wave
#define NCHUNK    (L_SEQ / CHUNK)            // 256
__global__ void scan_kernel(const float* __restrict__ xf, const float* __restrict__ xdbl,
                            const float* __restrict__ dtw, const float* __restrict__ dtb,
                            const float* __restrict__ Alog, const float* __restrict__ Ds,
                            float* __restrict__ ydir) {
  __shared__ float sbuf[2][CHUNK_F];
  const int lane = threadIdx.x & 31;
  const int dp = blockIdx.x;                 // 0..95
  const int k = blockIdx.y;                  // 0..3
  const int b = blockIdx.z;                  // 0..1
  const int d = dp * 2 + (lane >> 4);
  const int n = lane & 15;
  const int kd = k * D_INNER + d;

  const float A_dn = -__expf(Alog[kd * D_STATE + n]);
  const float Dd = Ds[kd];
  const float bias = dtb[kd];
  float w[DT_RANK];
#pragma unroll
  for (int r = 0; r < DT_RANK; ++r) w[r] = dtw[kd * DT_RANK + r];

  const char* xd = (const char*)(xdbl + (size_t)(b * K_DIRS + k) * L_SEQ * C_PAD);
  const float* uf = xf + (size_t)(b * D_INNER + d) * L_SEQ;
  float* yp = ydir + ((size_t)(b * K_DIRS + k) * D_INNER + d) * L_SEQ;

  const unsigned laneB = (unsigned)lane * 16u;
  const unsigned ldsA[2] = {(unsigned)(uintptr_t)&sbuf[0][0] + laneB,
                            (unsigned)(uintptr_t)&sbuf[1][0] + laneB};

  auto issue = [&](int c) {                  // async-copy one 3072B chunk into LDS
    const unsigned dst = ldsA[c & 1];
    const char* src = xd + (size_t)c * (CHUNK_F * 4) + laneB;
#pragma unroll
    for (int j = 0; j < 6; ++j)
      asm volatile("global_load_async_to_lds_b128 %0, %1, off offset:%2"
                   :: "v"(dst), "v"(src), "i"(j * 512) : "memory");
  };

  issue(0);
  float h = 0.f;
  for (int c = 0; c < NCHUNK; ++c) {
    if (c + 1 < NCHUNK) {
      issue(c + 1);
      __builtin_prefetch(&uf[dir_pos(k, (c + 1) * CHUNK)], 0, 0);
      WAIT_ASYNC(6);                         // in-order: chunk c resident, c+1 in flight
    } else {
      WAIT_ASYNC(0);
    }
    const float* col = &sbuf[c & 1][0];
#pragma unroll 4
    for (int t = 0; t < CHUNK; ++t) {
      const int l = c * CHUNK + t;
      const float* cc = col + t * C_PAD;
      float Bn = cc[DT_RANK + n];
      float Cn = cc[DT_RANK + D_STATE + n];
      float dr = bias;
#pragma unroll
      for (int r = 0; r < DT_RANK; ++r) dr = fmaf(w[r], cc[r], dr);
      float delta = (dr > 20.f) ? dr : log1pf(__expf(dr));   // softplus
      int pos = dir_pos(k, l);
      float u = uf[pos];
      h = fmaf(h, __expf(delta * A_dn), delta * u * Bn);
      float y = h * Cn;
      y += __shfl_xor(y, 8, 16);
      y += __shfl_xor(y, 4, 16);
      y += __shfl_xor(y, 2, 16);
      y += __shfl_xor(y, 1, 16);
      if (n == 0) yp[pos] = fmaf(u, Dd, y);                  // already de-permuted
    }
  }
}

// ---------------- K5: 4-way merge + LayerNorm(192) + SiLU(z) gate ----------------
__global__ void merge_ln(const float* __restrict__ ydir, const float* __restrict__ xz,
                         const float* __restrict__ gamma, const float* __restrict__ beta,
                         float* __restrict__ yg) {
  const int bl = blockIdx.x;           // b*L + l
  const int d = threadIdx.x;           // 0..191 (6 waves)
  const int b = bl / L_SEQ, l = bl & (L_SEQ - 1);
  float y = 0.f;
#pragma unroll
  for (int k = 0; k < K_DIRS; ++k)
    y += ydir[((size_t)(b * K_DIRS + k) * D_INNER + d) * L_SEQ + l];

  float s = y, s2 = y * y;
#pragma unroll
  for (int off = 16; off > 0; off >>= 1) {
    s += __shfl_xor(s, off, 32);
    s2 += __shfl_xor(s2, off, 32);
  }
  __shared__ float ws1[6], ws2[6];
  __shared__ float s_mu, s_rstd;
  if ((threadIdx.x & 31) == 0) { ws1[threadIdx.x >> 5] = s; ws2[threadIdx.x >> 5] = s2; }
  __syncthreads();
  if (threadIdx.x == 0) {
    float a = 0.f, q = 0.f;
#pragma unroll
    for (int i = 0; i < 6; ++i) { a += ws1[i]; q += ws2[i]; }
    float mu = a / (float)D_INNER;
    float var = q / (float)D_INNER - mu * mu;
    s_mu = mu;
    s_rstd = rsqrtf(var + LN_EPS);
  }
  __syncthreads();
  float yn = (y - s_mu) * s_rstd * gamma[d] + beta[d];
  float z = xz[(size_t)bl * (2 * D_INNER) + D_INNER + d];
  float g = z / (1.f + __expf(-z));
  yg[(size_t)bl * D_INNER + d] = yn * g;
}

// ---------------- host side ----------------
extern "C" void kernel_launch(void* const* d_in, const int* in_sizes, int n_in,
                              void* d_out, int out_size, void* d_ws, size_t ws_size,
                              hipStream_t stream) {
  const float* x        = (const float*)d_in[0];   // (B,H,W,96)
  const float* in_proj  = (const float*)d_in[1];   // (384,96)
  const float* conv_w   = (const float*)d_in[2];   // (192,1,3,3)
  const float* conv_b   = (const float*)d_in[3];   // (192)
  const float* xpw      = (const float*)d_in[4];   // (4,38,192)
  const float* dtw      = (const float*)d_in[5];   // (4,192,6)
  const float* dtb      = (const float*)d_in[6];   // (4,192)
  const float* Alog     = (const float*)d_in[7];   // (4,192,16)
  const float* Ds       = (const float*)d_in[8];   // (4,192)
  const float* ln_g     = (const float*)d_in[9];   // (192)
  const float* ln_b     = (const float*)d_in[10];  // (192)
  const float* out_proj = (const float*)d_in[11];  // (96,192)
  float* out = (float*)d_out;

  const int M = B_SZ * L_SEQ;                      // 8192 tokens
  float* ws = (float*)d_ws;
  float* xz   = ws;                                  size_t o = (size_t)M * 2 * D_INNER;
  float* xf   = ws + o;                              o += (size_t)B_SZ * D_INNER * L_SEQ;
  float* xfT  = ws + o;                              o += (size_t)B_SZ * L_SEQ * D_INNER;
  float* xdbl = ws + o;                              o += (size_t)B_SZ * K_DIRS * C_PAD * L_SEQ;
  float* ydir = ws + o;                              o += (size_t)B_SZ * K_DIRS * D_INNER * L_SEQ;
  float* yg   = ws + o;                              o += (size_t)M * D_INNER;
  float* xpwp = ws + o;

  // 0) pad x_proj weights to (4,48,192)
  pad_w<<<(K_DIRS * C_PAD * D_INNER + 255) / 256, 256, 0, stream>>>(xpw, xpwp);
  // 1) xz = x @ in_proj^T            (8192 x 384, K=96), 16x64 strip per wave
  gemm_nt_wmma<4><<<dim3(M / 16, (2 * D_INNER) / 64), 32, 0, stream>>>(
      x, in_proj, xz, M, 2 * D_INNER, D_MODEL);
  // 2) depthwise conv + SiLU -> xf (B,DI,L) and xfT (B,L,DI)
  conv_silu<<<(B_SZ * D_INNER * L_SEQ + 255) / 256, 256, 0, stream>>>(
      xz, conv_w, conv_b, xf, xfT);
  // 3) x_dbl per direction, column-major (B*K, L, 48), K=192
  xdbl_wmma<<<dim3(L_SEQ / 64, C_PAD / 16, B_SZ * K_DIRS), 32, 0, stream>>>(
      xfT, xpwp, xdbl);
  // 4) selective scan: 768 waves, async-LDS double buffered
  scan_kernel<<<dim3(D_INNER / 2, K_DIRS, B_SZ), 32, 0, stream>>>(
      xf, xdbl, dtw, dtb, Alog, Ds, ydir);
  // 5) merge + LN + gate
  merge_ln<<<M, D_INNER, 0, stream>>>(ydir, xz, ln_g, ln_b, yg);
  // 6) out = yg @ out_proj^T         (8192 x 96, K=192), 16x48 strip per wave
  gemm_nt_wmma<3><<<dim3(M / 16, D_MODEL / 48), 32, 0, stream>>>(
      yg, out_proj, out, M, D_MODEL, D_INNER);
}